// EquivariantProductBasisBlock_23175643529299
// MI455X (gfx1250) — compile-verified
//
#include <hip/hip_runtime.h>

// ---------------------------------------------------------------------------
// MACE EquivariantProductBasisBlock on MI455X (gfx1250), fp32 WMMA path.
//
// N=1024 atoms, C=64 channels, K=9 (irreps d=1,3,5), P3=4,P2=2,P1=1, E=10.
//
// Restructure: fold species-routed weights into the A operand so the U
// tensors become ONE shared B matrix (768 cols x 40 K-rows, L2-resident):
//   D[n, col(dd,a,b)] = sum_k A'[n,k] B[k,col]      (V_WMMA_F32_16X16X4_F32)
//   rows k=0..35 : A' = x[n,i]*w3[n,p]  (k = p*9+i), B = U3[dd,a,b,i,p]
//   rows 36..37  : A' = w2[n,p],                    B = U2[dd,a,(b as i),p]
//   rows 38..39  : zero pad (K=40 -> 10 WMMA K-steps)
// => D is directly c2[n,c,(dd,a),i=b]. Stage-2 (dot with x over b) is
// scattered from D lanes into LDS via ds_add_f32; stage-3 is tiny VALU.
// Final per-irrep channel mix [N*d x 64]*[64 x 64] + sc is a second WMMA GEMM.
//
// Data-movement: B stored TRANSPOSED (Bt[col][k]) so each lane's K-row pair
// is one aligned global_load_b64. O stored as Ot[n][k][c] and Wlin
// pre-transposed (WT[co][ci]) so the mix GEMM's A/B fetches are b64 too.
//
// WMMA f32 16x16x4 lane layout per CDNA5 ISA 7.12.2:
//   A (16x4):  lane L: M=L%16; v0 = K=(k0 + (L<16?0:2)), v1 = that+1
//   B (4x16):  lane L: N=L%16; v0 = K-row (k0 + (L<16?0:2)), v1 = that+1
//   C/D 16x16: lane L: N=L%16; VGPR r -> M = r + (L<16?0:8)
// ---------------------------------------------------------------------------

typedef float v2f __attribute__((ext_vector_type(2)));
typedef float v8f __attribute__((ext_vector_type(8)));

#define BCOLS 768            // per-irrep column blocks: [0,96) [96,352) [352,768)
#define BROWS 40             // K rows (38 used + 2 pad)
#define WT_OFFSET  131072    // byte offset of WT  (3 x 64 x 64 fp32)
#define O_OFFSET   196608    // byte offset of Ot ([N][9][64] fp32)

static __device__ __forceinline__ v8f wmma_f32_4(v2f a, v2f b, v8f c) {
  return __builtin_amdgcn_wmma_f32_16x16x4_f32(false, a, false, b,
                                               (short)0, c, false, false);
}

// ------------------- kernel 1a: build transposed B stack -------------------
__global__ void __launch_bounds__(256) build_B(
    const float* __restrict__ U3_0, const float* __restrict__ U2_0,
    const float* __restrict__ U3_1, const float* __restrict__ U2_1,
    const float* __restrict__ U3_2, const float* __restrict__ U2_2,
    float* __restrict__ Bt) {
  int idx = blockIdx.x * blockDim.x + threadIdx.x;
  if (idx >= BCOLS * BROWS) return;
  int col = idx / BROWS, k = idx % BROWS;
  int r, lo;
  if (col < 96)       { r = 0; lo = col; }
  else if (col < 352) { r = 1; lo = col - 96; }
  else                { r = 2; lo = col - 352; }
  int d = (r == 0) ? 1 : (r == 1 ? 3 : 5);
  const float* U3 = (r == 0) ? U3_0 : (r == 1 ? U3_1 : U3_2);
  const float* U2 = (r == 0) ? U2_0 : (r == 1 ? U2_1 : U2_2);
  float v = 0.f;
  if (lo < d * 81) {
    int dd = lo / 81, rem = lo % 81, a = rem / 9, b = rem % 9;
    if (k < 36) {                       // U3[dd,a,b,i,p], p fastest
      int p = k / 9, i = k % 9;
      v = U3[(((dd * 9 + a) * 9 + b) * 9 + i) * 4 + p];
    } else if (k < 38) {                // U2[dd,a,i(=b),p]
      int p = k - 36;
      v = U2[((dd * 9 + a) * 9 + b) * 2 + p];
    }
  }
  Bt[col * BROWS + k] = v;              // transposed: K contiguous per column
}

// ---------------------- kernel 1b: transpose Wlin x3 -----------------------
__global__ void __launch_bounds__(256) build_WT(
    const float* __restrict__ Wl0, const float* __restrict__ Wl1,
    const float* __restrict__ Wl2, float* __restrict__ WT) {
  int idx = blockIdx.x * blockDim.x + threadIdx.x;
  if (idx >= 3 * 4096) return;
  int r = idx >> 12, t = idx & 4095;
  int ci = t >> 6, co = t & 63;
  const float* Wl = (r == 0) ? Wl0 : (r == 1 ? Wl1 : Wl2);
  WT[r * 4096 + co * 64 + ci] = Wl[ci * 64 + co];
}

// ----------------------- kernel 2: main contraction ------------------------
template <int R>
__device__ __forceinline__ void contract_irrep(
    const float* __restrict__ W3r, const float* __restrict__ W2r,
    const float* __restrict__ Bt, const float* __restrict__ x,
    int e, int c, int half, int lm,
    const float (*xs)[9], float (*o2)[84]) {
  constexpr int D     = (R == 0) ? 1 : ((R == 1) ? 3 : 5);
  constexpr int COFF  = (R == 0) ? 0 : ((R == 1) ? 96 : 352);
  constexpr int CT    = (R == 0) ? 6 : ((R == 1) ? 16 : 26);
  constexpr int DAOFF = (R == 0) ? 0 : ((R == 1) ? 9 : 36);

  float w3[4], w2[2];
#pragma unroll
  for (int p = 0; p < 4; ++p) w3[p] = W3r[e * 256 + p * 64 + c];
#pragma unroll
  for (int p = 0; p < 2; ++p) w2[p] = W2r[e * 128 + p * 64 + c];

  // A' values this lane feeds (compile-time k after unroll -> no scratch)
  auto Aval = [&](int k) -> float {
    return (k < 36) ? x[k % 9] * w3[k / 9] : ((k < 38) ? w2[k - 36] : 0.f);
  };
  float aA[10], aB[10];
#pragma unroll
  for (int s = 0; s < 10; ++s) {
    float v0 = Aval(4 * s + 0), v1 = Aval(4 * s + 1);
    float v2 = Aval(4 * s + 2), v3 = Aval(4 * s + 3);
    aA[s] = half ? v2 : v0;
    aB[s] = half ? v3 : v1;
  }

  for (int ct = 0; ct < CT; ++ct) {
    int colb = COFF + ct * 16 + lm;
    const float* Bcol = Bt + colb * BROWS + half * 2;  // K pair base (8B aligned)
    v8f acc = {0.f, 0.f, 0.f, 0.f, 0.f, 0.f, 0.f, 0.f};
#pragma unroll
    for (int s = 0; s < 10; ++s) {
      v2f a; a.x = aA[s]; a.y = aB[s];
      v2f b = *(const v2f*)(Bcol + 4 * s);             // one global_load_b64
      acc = wmma_f32_4(a, b, acc);
    }
    // stage 2: out2[m, da] += c2[m, da, b] * x[m, b]
    int cl = ct * 16 + lm;              // local col within irrep = dd*81+a*9+b
    if (cl < D * 81) {
      int da = DAOFF + cl / 9;
      int b9 = cl - (cl / 9) * 9;
      float xv[8];
#pragma unroll
      for (int rr = 0; rr < 8; ++rr) xv[rr] = xs[rr + half * 8][b9];
#pragma unroll
      for (int rr = 0; rr < 8; ++rr)
        atomicAdd(&o2[rr + half * 8][da], acc[rr] * xv[rr]);
    }
  }
}

__global__ void __launch_bounds__(128) contract(
    const float* __restrict__ xin, const int* __restrict__ species,
    const float* __restrict__ W3_0, const float* __restrict__ W2_0,
    const float* __restrict__ W1_0, const float* __restrict__ U1_0,
    const float* __restrict__ W3_1, const float* __restrict__ W2_1,
    const float* __restrict__ W1_1, const float* __restrict__ U1_1,
    const float* __restrict__ W3_2, const float* __restrict__ W2_2,
    const float* __restrict__ W1_2, const float* __restrict__ U1_2,
    const float* __restrict__ Bt, float* __restrict__ Ot) {
  __shared__ float xs[4][16][9];
  __shared__ float o2[4][16][84];

  int w = threadIdx.x >> 5;
  int lane = threadIdx.x & 31;
  int half = lane >> 4, lm = lane & 15;
  int unit = blockIdx.x * 4 + w;        // 0..4095 = (atom_tile, channel)
  int tile = unit >> 6, c = unit & 63;
  int n0 = tile * 16;
  int nA = n0 + lm;

  float x[9];
#pragma unroll
  for (int i = 0; i < 9; ++i) x[i] = xin[(nA * 64 + c) * 9 + i];
  if (half == 0) {
#pragma unroll
    for (int i = 0; i < 9; ++i) xs[w][lm][i] = x[i];
  }
  int e = species[nA];

  // zero stage-2 accumulators (per-wave region)
  float* o2flat = &o2[w][0][0];
  for (int t = lane; t < 16 * 84; t += 32) o2flat[t] = 0.f;
  __syncthreads();

  contract_irrep<0>(W3_0, W2_0, Bt, x, e, c, half, lm, xs[w], o2[w]);
  contract_irrep<1>(W3_1, W2_1, Bt, x, e, c, half, lm, xs[w], o2[w]);
  contract_irrep<2>(W3_2, W2_2, Bt, x, e, c, half, lm, xs[w], o2[w]);
  __syncthreads();

  // stage 3: out1[dd] = sum_i (U1[dd,i]*w1 + out2[dd,i]) * x[i]
  if (half == 0) {
    int m = lm;
    float o[9];
    float w1_0 = W1_0[e * 64 + c];
    float w1_1 = W1_1[e * 64 + c];
    float w1_2 = W1_2[e * 64 + c];
    {
      float s = 0.f;
#pragma unroll
      for (int i = 0; i < 9; ++i)
        s += (U1_0[i] * w1_0 + o2[w][m][i]) * x[i];
      o[0] = s;
    }
#pragma unroll
    for (int dd = 0; dd < 3; ++dd) {
      float s = 0.f;
#pragma unroll
      for (int i = 0; i < 9; ++i)
        s += (U1_1[dd * 9 + i] * w1_1 + o2[w][m][9 + dd * 9 + i]) * x[i];
      o[1 + dd] = s;
    }
#pragma unroll
    for (int dd = 0; dd < 5; ++dd) {
      float s = 0.f;
#pragma unroll
      for (int i = 0; i < 9; ++i)
        s += (U1_2[dd * 9 + i] * w1_2 + o2[w][m][36 + dd * 9 + i]) * x[i];
      o[4 + dd] = s;
    }
    // Ot[n][k][c] layout: channel contiguous for the mix GEMM's A operand
#pragma unroll
    for (int k = 0; k < 9; ++k) Ot[(n0 + m) * 576 + k * 64 + c] = o[k];
  }
}

// ----------------- kernel 3: per-irrep channel mix + residual ---------------
__global__ void __launch_bounds__(128) mix(
    const float* __restrict__ Ot, const float* __restrict__ sc,
    const float* __restrict__ WT, float* __restrict__ out) {
  int w = threadIdx.x >> 5, lane = threadIdx.x & 31;
  int half = lane >> 4, lm = lane & 15;
  int unit = blockIdx.x * 4 + w;        // 0..2303
  int r, u;
  if (unit < 256)       { r = 0; u = unit; }
  else if (unit < 1024) { r = 1; u = unit - 256; }
  else                  { r = 2; u = unit - 1024; }
  int rt = u >> 2, ct = u & 3;
  int D = (r == 0) ? 1 : (r == 1 ? 3 : 5);
  int koff = (r == 0) ? 0 : (r == 1 ? 1 : 4);
  const float* Wr = WT + r * 4096;      // WT[co][ci]

  int co = ct * 16 + lm;
  int Rbase = rt * 16;
  int Ra = Rbase + lm;                  // A row this lane feeds
  int na = Ra / D, dda = Ra % D;
  const float* Arow = Ot + na * 576 + (koff + dda) * 64 + half * 2;  // + ci
  const float* Brow = Wr + co * 64 + half * 2;

  v8f acc = {0.f, 0.f, 0.f, 0.f, 0.f, 0.f, 0.f, 0.f};
#pragma unroll
  for (int s = 0; s < 16; ++s) {
    v2f a = *(const v2f*)(Arow + 4 * s);               // global_load_b64
    v2f b = *(const v2f*)(Brow + 4 * s);               // global_load_b64
    acc = wmma_f32_4(a, b, acc);
  }
#pragma unroll
  for (int rr = 0; rr < 8; ++rr) {
    int Rrow = Rbase + rr + half * 8;
    int n = Rrow / D, dd = Rrow % D;
    int idx = n * 576 + co * 9 + (koff + dd);
    out[idx] = acc[rr] + sc[idx];
  }
}

// ------------------------------- launcher ----------------------------------
extern "C" void kernel_launch(void* const* d_in, const int* in_sizes, int n_in,
                              void* d_out, int out_size, void* d_ws,
                              size_t ws_size, hipStream_t stream) {
  const float* xin = (const float*)d_in[0];
  const float* sc  = (const float*)d_in[1];
  const int* species = (const int*)d_in[2];
  const float *U3[3], *U2[3], *U1[3], *W3[3], *W2[3], *W1[3], *Wl[3];
  for (int r = 0; r < 3; ++r) {
    U3[r] = (const float*)d_in[3 + 7 * r + 0];
    U2[r] = (const float*)d_in[3 + 7 * r + 1];
    U1[r] = (const float*)d_in[3 + 7 * r + 2];
    W3[r] = (const float*)d_in[3 + 7 * r + 3];
    W2[r] = (const float*)d_in[3 + 7 * r + 4];
    W1[r] = (const float*)d_in[3 + 7 * r + 5];
    Wl[r] = (const float*)d_in[3 + 7 * r + 6];
  }
  float* Bt = (float*)d_ws;                            // 768*40*4 = 120 KB
  float* WT = (float*)((char*)d_ws + WT_OFFSET);       // 3*64*64*4 = 48 KB
  float* Ot = (float*)((char*)d_ws + O_OFFSET);        // [1024][9][64] fp32

  build_B<<<(BCOLS * BROWS + 255) / 256, 256, 0, stream>>>(
      U3[0], U2[0], U3[1], U2[1], U3[2], U2[2], Bt);
  build_WT<<<(3 * 4096 + 255) / 256, 256, 0, stream>>>(
      Wl[0], Wl[1], Wl[2], WT);
  contract<<<1024, 128, 0, stream>>>(
      xin, species,
      W3[0], W2[0], W1[0], U1[0],
      W3[1], W2[1], W1[1], U1[1],
      W3[2], W2[2], W1[2], U1[2],
      Bt, Ot);
  mix<<<576, 128, 0, stream>>>(Ot, sc, WT, (float*)d_out);
}